// SoftTopKLoss_49512382988774
// MI455X (gfx1250) — compile-verified
//
#include <hip/hip_runtime.h>
#include <math.h>

typedef __attribute__((ext_vector_type(2))) float v2f;
typedef __attribute__((ext_vector_type(8))) float v8f;

#define BATCH 256
#define NPOS 4
#define NNEG 2048
#define DIM 512
#define NITEMS (NPOS + NNEG)   // 2052
#define EPSV 1e-8f
#define INV_T 10.0f            // 1/TEMPERATURE
#define DECAY_OVER_K (2.0f / 3.0f)
#define POS_W 0.3f

// One block (256 threads = 8 waves, wave32) per batch row.
__global__ __launch_bounds__(256) void stk_main(
    const float* __restrict__ anchor,
    const float* __restrict__ positives,
    const float* __restrict__ negatives,
    float* __restrict__ partial)
{
    __shared__ __align__(16) float lds_anchor[DIM];
    __shared__ __align__(16) float lds_dot[NITEMS];   // dots, later sims
    __shared__ __align__(16) float lds_nrm[NITEMS];   // squared norms
    __shared__ float lds_redf[8];
    __shared__ int   lds_redi[NPOS * 8];
    __shared__ float lds_sc[2];                       // [0]=inv|a|, [1]=max sim

    const int b    = blockIdx.x;
    const int tid  = threadIdx.x;
    const int lane = tid & 31;
    const int wave = tid >> 5;
    const int half = lane >> 4;    // 0: K=0,1  1: K=2,3 (A-fragment half)
    const int mrow = lane & 15;    // item row within 16-item group

    // ---- Phase 0: stage anchor in LDS, compute 1/max(|a|,eps) ----
    const float* ap = anchor + (size_t)b * DIM;
    float ssa = 0.0f;
    for (int i = tid; i < DIM; i += 256) {
        float v = ap[i];
        lds_anchor[i] = v;
        ssa += v * v;
    }
    for (int off = 16; off; off >>= 1) ssa += __shfl_xor(ssa, off, 32);
    if (lane == 0) lds_redf[wave] = ssa;
    __syncthreads();
    if (tid == 0) {
        float s = 0.0f;
        for (int w = 0; w < 8; ++w) s += lds_redf[w];
        lds_sc[0] = 1.0f / fmaxf(sqrtf(s), EPSV);
    }
    __syncthreads();

    // ---- Phase 1a: positives (waves 0..3, one each): plain dot + norm ----
    if (wave < NPOS) {
        const float* pp = positives + ((size_t)b * NPOS + wave) * DIM;
        float dot = 0.0f, ss = 0.0f;
        for (int i = lane; i < DIM; i += 32) {
            float v = pp[i];
            dot += v * lds_anchor[i];
            ss  += v * v;
        }
        for (int off = 16; off; off >>= 1) {
            dot += __shfl_xor(dot, off, 32);
            ss  += __shfl_xor(ss,  off, 32);
        }
        if (lane == 0) { lds_dot[wave] = dot; lds_nrm[wave] = ss; }
    }

    // ---- Phase 1b: negatives via V_WMMA_F32_16X16X4_F32 ----
    // A = 16 negatives x 4 dims (lane half supplies K={0,1} or {2,3} as float2)
    // B = anchor chunk broadcast across all 16 columns (uniform LDS read per half)
    // After 128 chunks, accumulator column N=0 (lanes 0 & 16) = the 16 dots.
    for (int g = wave; g < NNEG / 16; g += 8) {
        const float* rowp = negatives
            + ((size_t)b * NNEG + (size_t)g * 16 + mrow) * DIM + half * 2;
        v8f acc = {0.f, 0.f, 0.f, 0.f, 0.f, 0.f, 0.f, 0.f};
        float ss = 0.0f;
        #pragma unroll 4
        for (int c = 0; c < DIM / 4; ++c) {
            v2f a2 = *(const v2f*)(rowp + 4 * c);
            v2f b2 = *(const v2f*)(&lds_anchor[4 * c + half * 2]);
            ss += a2[0] * a2[0] + a2[1] * a2[1];
            acc = __builtin_amdgcn_wmma_f32_16x16x4_f32(
                false, a2, false, b2, (short)0, acc, false, false);
        }
        // item mrow's |x|^2 = this half's partial + other half's partial
        float ss_o = __shfl_xor(ss, 16, 32);
        if (half == 0) lds_nrm[NPOS + g * 16 + mrow] = ss + ss_o;
        // extract column N=0: lane 0 holds M=0..7, lane 16 holds M=8..15
        if (mrow == 0) {
            int mb = NPOS + g * 16 + half * 8;
            #pragma unroll
            for (int r = 0; r < 8; ++r) lds_dot[mb + r] = acc[r];
        }
    }
    __syncthreads();

    // ---- Phase 2: finalize sims = dot * inv|a| * inv|x| ----
    const float inv_a = lds_sc[0];
    for (int i = tid; i < NITEMS; i += 256) {
        float inv_x = 1.0f / fmaxf(sqrtf(lds_nrm[i]), EPSV);
        lds_dot[i] = lds_dot[i] * inv_a * inv_x;
    }
    __syncthreads();

    // ---- Phase 3: block max over sims ----
    float mx = -3.402823466e38f;
    for (int i = tid; i < NITEMS; i += 256) mx = fmaxf(mx, lds_dot[i]);
    for (int off = 16; off; off >>= 1) mx = fmaxf(mx, __shfl_xor(mx, off, 32));
    if (lane == 0) lds_redf[wave] = mx;
    __syncthreads();
    if (tid == 0) {
        float m = lds_redf[0];
        for (int w = 1; w < 8; ++w) m = fmaxf(m, lds_redf[w]);
        lds_sc[1] = m;
    }
    __syncthreads();

    // ---- Phase 4: sum of exp + rank counts for the 4 positives ----
    const float smax = lds_sc[1];
    float sp0 = lds_dot[0], sp1 = lds_dot[1], sp2 = lds_dot[2], sp3 = lds_dot[3];
    float se = 0.0f;
    int c0 = 0, c1 = 0, c2 = 0, c3 = 0;
    for (int i = tid; i < NITEMS; i += 256) {
        float s = lds_dot[i];
        se += expf((s - smax) * INV_T);
        c0 += (s > sp0); c1 += (s > sp1); c2 += (s > sp2); c3 += (s > sp3);
    }
    for (int off = 16; off; off >>= 1) {
        se += __shfl_xor(se, off, 32);
        c0 += __shfl_xor(c0, off, 32);
        c1 += __shfl_xor(c1, off, 32);
        c2 += __shfl_xor(c2, off, 32);
        c3 += __shfl_xor(c3, off, 32);
    }
    if (lane == 0) {
        lds_redf[wave] = se;
        lds_redi[0 * 8 + wave] = c0;
        lds_redi[1 * 8 + wave] = c1;
        lds_redi[2 * 8 + wave] = c2;
        lds_redi[3 * 8 + wave] = c3;
    }
    __syncthreads();

    // ---- Phase 5: combine (thread 0) ----
    if (tid == 0) {
        float sumexp = 0.0f;
        int cnt[NPOS] = {0, 0, 0, 0};
        for (int w = 0; w < 8; ++w) {
            sumexp += lds_redf[w];
            for (int p = 0; p < NPOS; ++p) cnt[p] += lds_redi[p * 8 + w];
        }
        float simp[NPOS] = {sp0, sp1, sp2, sp3};
        float pwp = 0.0f, psum = 0.0f;
        for (int p = 0; p < NPOS; ++p) {
            // stable-sort tie handling: earlier equal positives rank ahead
            int rank = cnt[p];
            for (int q = 0; q < p; ++q) rank += (simp[q] == simp[p]);
            float prob = expf((simp[p] - smax) * INV_T) / sumexp;
            float w = expf(-DECAY_OVER_K * (float)rank);
            pwp  += prob * w;
            psum += simp[p];
        }
        // per-row contribution to the global scalar loss
        partial[b] = -pwp / (float)BATCH
                     + POS_W / (float)BATCH
                     - POS_W * psum / (float)(BATCH * NPOS);
    }
}

// Deterministic final sum of the 256 per-row partials.
__global__ __launch_bounds__(256) void stk_reduce(const float* __restrict__ partial,
                                                  float* __restrict__ out)
{
    __shared__ float s[256];
    int tid = threadIdx.x;
    s[tid] = partial[tid];
    __syncthreads();
    for (int k = 128; k > 0; k >>= 1) {
        if (tid < k) s[tid] += s[tid + k];
        __syncthreads();
    }
    if (tid == 0) out[0] = s[0];
}

extern "C" void kernel_launch(void* const* d_in, const int* in_sizes, int n_in,
                              void* d_out, int out_size, void* d_ws, size_t ws_size,
                              hipStream_t stream) {
    (void)in_sizes; (void)n_in; (void)out_size; (void)ws_size;
    const float* anchor    = (const float*)d_in[0];
    const float* positives = (const float*)d_in[1];
    const float* negatives = (const float*)d_in[2];
    float* partial = (float*)d_ws;           // 256 floats of scratch
    float* out     = (float*)d_out;

    stk_main<<<BATCH, 256, 0, stream>>>(anchor, positives, negatives, partial);
    stk_reduce<<<1, 256, 0, stream>>>(partial, out);
}